// reservoir_net_74182675136948
// MI455X (gfx1250) — compile-verified
//
#include <hip/hip_runtime.h>
#include <hip/hip_bf16.h>
#include <math.h>

// Problem constants (match reference)
#define BB 32
#define TT 512
#define NN 1024
#define RR 4
#define II 8
#define DT_ 0.1f

typedef __attribute__((ext_vector_type(2))) float v2f;
typedef __attribute__((ext_vector_type(8))) float v8f;

// ---------------- init kernels ----------------
__global__ void init_p_kernel(float* __restrict__ P) {
    size_t total = (size_t)BB * NN * NN;
    for (size_t idx = (size_t)blockIdx.x * blockDim.x + threadIdx.x; idx < total;
         idx += (size_t)gridDim.x * blockDim.x) {
        size_t w = idx & ((size_t)NN * NN - 1);     // within-batch index (N*N is pow2)
        size_t row = w >> 10;                        // /N
        size_t col = w & (NN - 1);                   // %N
        P[idx] = (row == col) ? 0.1f : 0.0f;
    }
}

__global__ void init_zero_kernel(float* __restrict__ p, size_t count) {
    for (size_t idx = (size_t)blockIdx.x * blockDim.x + threadIdx.x; idx < count;
         idx += (size_t)gridDim.x * blockDim.x) {
        p[idx] = 0.0f;
    }
}

// ---------------- K1: fused P rank-1 update (WMMA) + P@r + J@r + rPr ----------------
// grid = (N/16, B), block = 256 (8 waves). Wave handles 8 of the 64 column tiles
// of a 16-row strip of P. The rank-1 update D = A*B + C is one
// v_wmma_f32_16x16x4_f32 per tile (A col0 = -c*Pr rows, B row0 = Pr cols,
// C = P tile); the updated tile is dotted with r directly from the D registers,
// so P is read exactly once and written exactly once per time step.
__global__ __launch_bounds__(256)
void k1_matvec(float* __restrict__ P, const float* __restrict__ J,
               const float* __restrict__ r, const float* __restrict__ PrPrev,
               float* __restrict__ PrCur, float* __restrict__ Jr,
               float* rPrBuf, int t) {
    const int b    = blockIdx.y;
    const int rt   = blockIdx.x;          // row tile 0..63
    const int tid  = threadIdx.x;
    const int wave = tid >> 5;
    const int lane = tid & 31;
    const int half = lane >> 4;           // 0: lanes 0-15, 1: lanes 16-31
    const int l    = lane & 15;

    __shared__ __align__(16) float ldsR[NN];
    __shared__ __align__(16) float ldsPr[NN];
    __shared__ float prPart[8][16];
    __shared__ float jrPart[16][17];
    __shared__ float red16[16];

    float*       Pb = P + (size_t)b * NN * NN;
    const float* Jb = J + (size_t)b * NN * NN;

    // stage r_t and Pr_{t-1} (zeros at t==0) into LDS; 256 threads * float4 = 1024
    {
        const float4* r4 = (const float4*)(r + (size_t)b * NN);
        ((float4*)ldsR)[tid] = r4[tid];
        if (t > 0) {
            const float4* p4 = (const float4*)(PrPrev + (size_t)b * NN);
            ((float4*)ldsPr)[tid] = p4[tid];
        } else {
            ((float4*)ldsPr)[tid] = make_float4(0.f, 0.f, 0.f, 0.f);
        }
    }
    __syncthreads();

    float cPrev = 0.0f;
    if (t > 0) cPrev = 1.0f / (1.0f + rPrBuf[(size_t)(t - 1) * BB + b]);

    const int rowbase = rt * 16;

    // A fragment (16x4 f32): VGPR0 lanes0-15 = (M=lane,K=0); all other K = 0
    const float hsel = (half == 0) ? 1.0f : 0.0f;
    v2f A; A.x = -cPrev * ldsPr[rowbase + l] * hsel; A.y = 0.0f;

    float acc[8];
#pragma unroll
    for (int v = 0; v < 8; ++v) acc[v] = 0.0f;

    // fixed trip count: wave w covers column tiles w, w+8, ..., w+56
#pragma unroll 2
    for (int j = 0; j < 8; ++j) {
        const int colbase = (j * 8 + wave) * 16;
        // B fragment (4x16 f32): VGPR0 lanes0-15 = (K=0, N=lane); rest 0
        v2f Bf; Bf.x = ldsPr[colbase + l] * hsel; Bf.y = 0.0f;

        // C/D layout: VGPR v holds rows M=v (lanes 0-15) and M=v+8 (lanes 16-31)
        size_t base = (size_t)(rowbase + half * 8) * NN + colbase + l;
        v8f C;
#pragma unroll
        for (int v = 0; v < 8; ++v) C[v] = Pb[base + (size_t)v * NN];

        v8f D = __builtin_amdgcn_wmma_f32_16x16x4_f32(false, A, false, Bf,
                                                      (short)0, C, false, false);

        // store updated P tile
#pragma unroll
        for (int v = 0; v < 8; ++v) Pb[base + (size_t)v * NN] = D[v];

        // dot updated tile with r (from registers; no P re-read)
        float rv = ldsR[colbase + l];
#pragma unroll
        for (int v = 0; v < 8; ++v) acc[v] += D[v] * rv;
    }

    // reduce across the 16 lanes of each half; xor masks 8..1 never cross
    // the 16-lane boundary, so full-width shuffles are safe and lower cleanly
#pragma unroll
    for (int v = 0; v < 8; ++v) {
        float a = acc[v];
        a += __shfl_xor(a, 8);
        a += __shfl_xor(a, 4);
        a += __shfl_xor(a, 2);
        a += __shfl_xor(a, 1);
        acc[v] = a;
    }
    if (l == 0) {
#pragma unroll
        for (int v = 0; v < 8; ++v) prPart[wave][half * 8 + v] = acc[v];
    }

    // J @ r for the same 16 rows: 16 threads per row, float4 loads
    {
        const int row = tid >> 4;
        const int s   = tid & 15;
        const float4* jrow = (const float4*)(Jb + (size_t)(rowbase + row) * NN);
        const float4* r4   = (const float4*)ldsR;
        float accj = 0.0f;
        for (int k = s; k < NN / 4; k += 16) {
            float4 jv = jrow[k];
            float4 rv = r4[k];
            accj += jv.x * rv.x + jv.y * rv.y + jv.z * rv.z + jv.w * rv.w;
        }
        jrPart[row][s] = accj;
    }
    __syncthreads();

    if (tid < 16) {
        float pr = 0.0f;
#pragma unroll
        for (int w = 0; w < 8; ++w) pr += prPart[w][tid];
        PrCur[(size_t)b * NN + rowbase + tid] = pr;

        float jr = 0.0f;
#pragma unroll
        for (int s = 0; s < 16; ++s) jr += jrPart[tid][s];
        Jr[(size_t)b * NN + rowbase + tid] = jr;

        red16[tid] = ldsR[rowbase + tid] * pr;   // partial r^T P r
    }
    __syncthreads();
    if (tid == 0) {
        float s = 0.0f;
#pragma unroll
        for (int i = 0; i < 16; ++i) s += red16[i];
        atomicAdd(&rPrBuf[(size_t)t * BB + b], s);
    }
}

// ---------------- K2: per-row state update ----------------
// grid = (N/256, B), block = 256
__global__ __launch_bounds__(256)
void k2_update(const float* __restrict__ inputs, const float* __restrict__ f,
               const float* __restrict__ wIn, const float* __restrict__ m,
               const float* __restrict__ rPrBuf, const float* __restrict__ PrCur,
               const float* __restrict__ Jr, float* __restrict__ rState,
               float* __restrict__ nState, float* zacc,
               float* __restrict__ allX, float* __restrict__ allZ, int t) {
    const int b   = blockIdx.y;
    const int tid = threadIdx.x;
    const int i   = blockIdx.x * 256 + tid;

    const float c = 1.0f / (1.0f + rPrBuf[(size_t)t * BB + b]);

    float z[RR], err[RR];
#pragma unroll
    for (int k = 0; k < RR; ++k) {
        z[k]   = zacc[((size_t)t * BB + b) * RR + k];
        err[k] = z[k] - f[((size_t)b * TT + t) * RR + k];
    }
    float u[II];
#pragma unroll
    for (int q = 0; q < II; ++q) u[q] = inputs[((size_t)b * TT + t) * II + q];

    float xv  = (t == 0) ? 0.0f : allX[((size_t)b * TT + (t - 1)) * NN + i];
    float sum = Jr[(size_t)b * NN + i];
#pragma unroll
    for (int k = 0; k < RR; ++k) sum += m[((size_t)b * NN + i) * RR + k] * z[k];
#pragma unroll
    for (int q = 0; q < II; ++q) sum += wIn[((size_t)b * NN + i) * II + q] * u[q];

    float xn = (1.0f - DT_) * xv + DT_ * sum;
    allX[((size_t)b * TT + t) * NN + i] = xn;
    float rn = tanhf(xn);
    rState[(size_t)b * NN + i] = rn;

    float pri = PrCur[(size_t)b * NN + i];
    float zc[RR];
#pragma unroll
    for (int k = 0; k < RR; ++k) {
        float nn = nState[((size_t)b * NN + i) * RR + k] - c * pri * err[k];
        nState[((size_t)b * NN + i) * RR + k] = nn;
        zc[k] = nn * rn;   // contribution to z_{t+1}
    }

    // block-reduce z_{t+1} contributions -> 4 atomics per block
    __shared__ float red[256];
#pragma unroll
    for (int k = 0; k < RR; ++k) {
        red[tid] = zc[k];
        __syncthreads();
        for (int s = 128; s > 0; s >>= 1) {
            if (tid < s) red[tid] += red[tid + s];
            __syncthreads();
        }
        if (tid == 0) atomicAdd(&zacc[(((size_t)t + 1) * BB + b) * RR + k], red[0]);
        __syncthreads();
    }

    if (blockIdx.x == 0 && tid < RR) allZ[((size_t)b * TT + t) * RR + tid] = z[tid];
}

// ---------------- host ----------------
extern "C" void kernel_launch(void* const* d_in, const int* in_sizes, int n_in,
                              void* d_out, int out_size, void* d_ws, size_t ws_size,
                              hipStream_t stream) {
    const float* inputs = (const float*)d_in[0];   // (B,T,I,1)
    const float* f      = (const float*)d_in[1];   // (B,T,R,1)
    const float* J      = (const float*)d_in[2];   // (B,N,N)
    const float* wIn    = (const float*)d_in[3];   // (B,N,I)
    const float* m      = (const float*)d_in[4];   // (B,N,R)

    float* allX = (float*)d_out;                         // (B,T,N)
    float* allZ = allX + (size_t)BB * TT * NN;           // (B,T,R)

    float* ws = (float*)d_ws;
    size_t off = 0;
    float* P      = ws + off; off += (size_t)BB * NN * NN;       // 128 MB
    float* rState = ws + off; off += (size_t)BB * NN;
    float* nState = ws + off; off += (size_t)BB * NN * RR;
    float* Pr0    = ws + off; off += (size_t)BB * NN;
    float* Pr1    = ws + off; off += (size_t)BB * NN;
    float* Jr     = ws + off; off += (size_t)BB * NN;
    float* rPr    = ws + off; off += (size_t)BB * TT;
    float* zacc   = ws + off; off += (size_t)BB * (TT + 1) * RR;

    // init: P = I/10 per batch; zero all accumulator/state arrays (contiguous)
    init_p_kernel<<<4096, 256, 0, stream>>>(P);
    size_t zcount = off - (size_t)BB * NN * NN;
    init_zero_kernel<<<1024, 256, 0, stream>>>(rState, zcount);

    for (int t = 0; t < TT; ++t) {
        const float* prPrev = (t & 1) ? Pr0 : Pr1;
        float*       prCur  = (t & 1) ? Pr1 : Pr0;
        dim3 g1(NN / 16, BB);
        k1_matvec<<<g1, 256, 0, stream>>>(P, J, rState, prPrev, prCur, Jr, rPr, t);
        dim3 g2(NN / 256, BB);
        k2_update<<<g2, 256, 0, stream>>>(inputs, f, wIn, m, rPr, prCur, Jr,
                                          rState, nState, zacc, allX, allZ, t);
    }
}